// UVCrossAttention_87737591923352
// MI455X (gfx1250) — compile-verified
//
#include <hip/hip_runtime.h>
#include <hip/hip_bf16.h>
#include <math.h>

typedef __attribute__((ext_vector_type(2))) float v2f;
typedef __attribute__((ext_vector_type(8))) float v8f;

#define NQ      1024
#define NV      1024
#define EMB     128
#define NDEPTH  128
#define NPOINTS 4

// ---------------------------------------------------------------------------
// Generic f32 WMMA GEMM: C[M,N] = A[M,K] @ B + bias + resid
//   transB==0: B stored [K,N] row-major
//   transB==1: B stored [N,K] row-major (logical B[k][n] = Bsrc[n*K+k])
// One wave computes one 16x16 C tile via V_WMMA_F32_16X16X4_F32, K-loop step 4.
// Block = 128 threads = 4 waves, each wave a different N-tile (N % 64 == 0).
// ---------------------------------------------------------------------------
__global__ __launch_bounds__(128)
void wmma_gemm_f32(const float* __restrict__ A, const float* __restrict__ B,
                   const float* __restrict__ bias, const float* __restrict__ resid,
                   float* __restrict__ C, int M, int N, int K, int transB)
{
    const int wid   = threadIdx.x >> 5;
    const int lane  = threadIdx.x & 31;
    const int half  = lane >> 4;     // 0: lanes 0-15, 1: lanes 16-31
    const int l16   = lane & 15;
    const int tileN = (blockIdx.x * 4 + wid) * 16;
    const int tileM = blockIdx.y * 16;

    const float* Arow = A + (size_t)(tileM + l16) * K;
    v8f c = {0.f, 0.f, 0.f, 0.f, 0.f, 0.f, 0.f, 0.f};

    if (transB) {
        const float* Brow = B + (size_t)(tileN + l16) * K;
        for (int k0 = 0; k0 < K; k0 += 4) {
            const int ka = k0 + 2 * half;          // K split across lane halves
            v2f a; a.x = Arow[ka]; a.y = Arow[ka + 1];
            v2f b; b.x = Brow[ka]; b.y = Brow[ka + 1];
            c = __builtin_amdgcn_wmma_f32_16x16x4_f32(false, a, false, b,
                                                      (short)0, c, false, false);
        }
    } else {
        for (int k0 = 0; k0 < K; k0 += 4) {
            const int ka = k0 + 2 * half;
            v2f a; a.x = Arow[ka]; a.y = Arow[ka + 1];
            v2f b;
            b.x = B[(size_t)ka * N + tileN + l16];
            b.y = B[(size_t)(ka + 1) * N + tileN + l16];
            c = __builtin_amdgcn_wmma_f32_16x16x4_f32(false, a, false, b,
                                                      (short)0, c, false, false);
        }
    }

    // Epilogue: C/D layout — VGPR j holds row (j + 8*half), column = lane%16.
    const int n  = tileN + l16;
    const float bv = bias ? bias[n] : 0.0f;
#pragma unroll
    for (int j = 0; j < 8; ++j) {
        const int m = tileM + j + 8 * half;
        float v = c[j] + bv;
        if (resid) v += resid[(size_t)m * N + n];
        C[(size_t)m * N + n] = v;
    }
}

// ---------------------------------------------------------------------------
// Per-query sampling kernel.
//   dot[q,v] = <key2[q], val[v]> is precomputed (WMMA GEMM). This kernel does,
//   per (q, d):  softmax over the 4 point logits, then
//   out_pre[q,d] = (1/128) * sum_p aw_p * bilinear(dot_row_q, x_p, y_p)
// One block per query q; dot row (4KB) staged in LDS; thread d = depth.
// ---------------------------------------------------------------------------
__global__ __launch_bounds__(128)
void uv_sample_kernel(const float* __restrict__ dotmat,   // [NQ, NV]
                      const float* __restrict__ off,      // [NQ, NDEPTH*NPOINTS*2]
                      const float* __restrict__ aw,       // [NQ, NDEPTH*NPOINTS]
                      const float* __restrict__ ref3d,    // [NQ*NDEPTH, 2]
                      const int*   __restrict__ sshapes,  // [2] = {H, W}
                      float*       __restrict__ out_pre)  // [NQ, NDEPTH]
{
    __shared__ float row[NV];
    const int q = blockIdx.x;
    const int t = threadIdx.x;

    for (int i = t; i < NV; i += 128) row[i] = dotmat[(size_t)q * NV + i];
    __syncthreads();

    const int H = sshapes[0];
    const int W = sshapes[1];
    const float fW = (float)W, fH = (float)H;

    const int d = t;

    // softmax over the 4 point logits (layout: q*512 + d*4 + p)
    const float* lg = aw + (size_t)q * (NDEPTH * NPOINTS) + d * NPOINTS;
    const float l0 = lg[0], l1 = lg[1], l2 = lg[2], l3 = lg[3];
    const float mx = fmaxf(fmaxf(l0, l1), fmaxf(l2, l3));
    float e[NPOINTS];
    e[0] = expf(l0 - mx); e[1] = expf(l1 - mx);
    e[2] = expf(l2 - mx); e[3] = expf(l3 - mx);
    const float inv = 1.0f / (e[0] + e[1] + e[2] + e[3]);

    const float rx = ref3d[((size_t)q * NDEPTH + d) * 2 + 0];
    const float ry = ref3d[((size_t)q * NDEPTH + d) * 2 + 1];
    const float* op = off + (size_t)q * (NDEPTH * NPOINTS * 2) + d * (NPOINTS * 2);

    float acc = 0.0f;
#pragma unroll
    for (int p = 0; p < NPOINTS; ++p) {
        const float x = (rx + op[p * 2 + 0] / fW) * fW - 0.5f;
        const float y = (ry + op[p * 2 + 1] / fH) * fH - 0.5f;
        const float x0f = floorf(x), y0f = floorf(y);
        const int ix0 = (int)x0f, iy0 = (int)y0f;
        const float wx1 = x - x0f, wy1 = y - y0f;
        const float wx0 = 1.0f - wx1, wy0 = 1.0f - wy1;

        float sp = 0.0f;
        { const int ix = ix0,     iy = iy0;     const float w = wx0 * wy0;
          if (ix >= 0 && ix < W && iy >= 0 && iy < H) sp += row[iy * W + ix] * w; }
        { const int ix = ix0 + 1, iy = iy0;     const float w = wx1 * wy0;
          if (ix >= 0 && ix < W && iy >= 0 && iy < H) sp += row[iy * W + ix] * w; }
        { const int ix = ix0,     iy = iy0 + 1; const float w = wx0 * wy1;
          if (ix >= 0 && ix < W && iy >= 0 && iy < H) sp += row[iy * W + ix] * w; }
        { const int ix = ix0 + 1, iy = iy0 + 1; const float w = wx1 * wy1;
          if (ix >= 0 && ix < W && iy >= 0 && iy < H) sp += row[iy * W + ix] * w; }

        acc += e[p] * inv * sp;
    }

    // mean over E (=128); mean over cams is a no-op (cams==1)
    out_pre[(size_t)q * NDEPTH + d] = acc * (1.0f / 128.0f);
}

// ---------------------------------------------------------------------------
extern "C" void kernel_launch(void* const* d_in, const int* in_sizes, int n_in,
                              void* d_out, int out_size, void* d_ws, size_t ws_size,
                              hipStream_t stream) {
    const float* query  = (const float*)d_in[0];   // [1,1024,128]
    const float* key    = (const float*)d_in[1];   // [1,1024,1,128] -> flat == key2
    const float* value  = (const float*)d_in[2];   // [1,1024,1,128] (flip over cams=1 is no-op)
    const float* ref3d  = (const float*)d_in[3];   // [1,131072,2]
    const int*   sshape = (const int*)d_in[4];     // [[H,W]]
    const float* W_off  = (const float*)d_in[5];   // [128,1024]
    const float* b_off  = (const float*)d_in[6];   // [1024]
    const float* W_attn = (const float*)d_in[7];   // [128,512]
    const float* b_attn = (const float*)d_in[8];   // [512]
    const float* W_v    = (const float*)d_in[9];   // [128,128]
    const float* b_v    = (const float*)d_in[10];  // [128]
    const float* W_o    = (const float*)d_in[11];  // [128,128]
    const float* b_o    = (const float*)d_in[12];  // [128]
    float* out = (float*)d_out;                    // [1,1024,128]

    float* ws  = (float*)d_ws;
    float* val = ws;                               // 1024*128
    float* off = val + (size_t)NV * EMB;           // 1024*1024
    float* aw  = off + (size_t)NQ * 1024;          // 1024*512
    float* dot = aw  + (size_t)NQ * 512;           // 1024*1024
    float* pre = dot + (size_t)NQ * NV;            // 1024*128

    const dim3 blk(128, 1, 1);

    // val = value @ W_v + b_v                        (M=1024, N=128,  K=128)
    wmma_gemm_f32<<<dim3(EMB / 64, NV / 16), blk, 0, stream>>>(
        value, W_v, b_v, nullptr, val, NV, EMB, EMB, 0);

    // off = query @ W_off + b_off                    (M=1024, N=1024, K=128)
    wmma_gemm_f32<<<dim3(1024 / 64, NQ / 16), blk, 0, stream>>>(
        query, W_off, b_off, nullptr, off, NQ, 1024, EMB, 0);

    // aw  = query @ W_attn + b_attn                  (M=1024, N=512,  K=128)
    wmma_gemm_f32<<<dim3(512 / 64, NQ / 16), blk, 0, stream>>>(
        query, W_attn, b_attn, nullptr, aw, NQ, 512, EMB, 0);

    // dot = key2 @ val^T                             (M=1024, N=1024, K=128, transB)
    wmma_gemm_f32<<<dim3(NV / 64, NQ / 16), blk, 0, stream>>>(
        key, val, nullptr, nullptr, dot, NQ, NV, EMB, 1);

    // per-(q,d) softmax + bilinear gather from LDS-staged dot row
    uv_sample_kernel<<<dim3(NQ, 1, 1), blk, 0, stream>>>(
        dot, off, aw, ref3d, sshape, pre);

    // out = pre @ W_o + b_o + query (residual)       (M=1024, N=128, K=128)
    wmma_gemm_f32<<<dim3(EMB / 64, NQ / 16), blk, 0, stream>>>(
        pre, W_o, b_o, query, out, NQ, EMB, NDEPTH, 0);
}